// Model_37890201485585
// MI455X (gfx1250) — compile-verified
//
#include <hip/hip_runtime.h>
#include <hip/hip_bf16.h>
#include <math.h>

// ---------------- types / WMMA helper (CDNA5 f32 16x16x4) ----------------
typedef float v8f __attribute__((ext_vector_type(8)));
typedef float v2f __attribute__((ext_vector_type(2)));

static __device__ __forceinline__ v8f wmma4(v2f a, v2f b, v8f c) {
    // D = A(16x4 f32) * B(4x16 f32) + C(16x16 f32)
    return __builtin_amdgcn_wmma_f32_16x16x4_f32(
        /*neg_a=*/false, a, /*neg_b=*/false, b,
        /*c_mod=*/(short)0, c, /*reuse_a=*/false, /*reuse_b=*/false);
}

static __device__ __forceinline__ v2f lds_v2(const float* p) {
    return *(const v2f*)p;   // caller guarantees 8B alignment
}

// CDNA5 async direct global->LDS copy (16B), tracked by ASYNCcnt.
// Low 32 bits of a generic pointer into the LDS aperture are the wave-relative
// LDS byte offset (ISA 10.2).
static __device__ __forceinline__ void async_ld16(void* lds, const void* g) {
    unsigned l = (unsigned)(unsigned long long)lds;
    asm volatile("global_load_async_to_lds_b128 %0, %1, off"
                 :: "v"(l), "v"(g) : "memory");
}
static __device__ __forceinline__ void wait_async0() {
    asm volatile("s_wait_asynccnt 0x0" ::: "memory");
}

#define NB   64
#define NC   321
#define NCP  336      // 21*16, padded channel count
#define NS   30
#define NW   24
#define NHID 16
#define SEQ  720

// ---------------- K0: build orthonormal DCT-II matrix M[24][24] ----------------
__global__ void k_initM(float* __restrict__ Mout) {
    const float PI = 3.14159265358979323846f;
    for (int i = threadIdx.x; i < NW * NW; i += 256) {
        int k = i / NW, w = i % NW;
        float s = (k == 0) ? sqrtf(1.0f / 24.0f) : sqrtf(2.0f / 24.0f);
        Mout[i] = s * cosf(PI * (2.0f * (float)w + 1.0f) * (float)k / 48.0f);
    }
}

// ---------------- K1: per-(b,c) mean over time ----------------
__global__ void k_mean(const float* __restrict__ x, float* __restrict__ meanp) {
    int b = blockIdx.x;
    int c = threadIdx.x;
    if (c >= NC) return;
    const float* p = x + (size_t)b * SEQ * NC + c;
    float s = 0.f;
    for (int t = 0; t < SEQ; ++t) s += p[(size_t)t * NC];
    meanp[b * NC + c] = s * (1.0f / 720.0f);
}

// ---------------- K2: DCT via WMMA: per (b,s), out[c,k] = sum_w (x-mean)[w,c]*M[k,w] ----------------
__global__ void k_dct(const float* __restrict__ x, const float* __restrict__ Mp,
                      const float* __restrict__ meanp, float* __restrict__ xd) {
    __shared__ __align__(16) float sx[NCP * 26];   // [c][w], stride 26 (even -> v2f loads)
    __shared__ __align__(16) float sM[NW * NW];
    __shared__ float smean[NC];

    int bid = blockIdx.x;
    int b = bid / NS, s = bid % NS;
    int tid = threadIdx.x;

    for (int i = tid; i < NW * NW; i += 256) sM[i] = Mp[i];
    for (int i = tid; i < NC;      i += 256) smean[i] = meanp[b * NC + i];
    __syncthreads();

    // stage x[b, s*24+w, c] transposed into sx[c][w], mean-subtracted; zero-pad c>=321
    const float* xsrc = x + ((size_t)b * SEQ + (size_t)s * NW) * NC;
    for (int i = tid; i < NW * NC; i += 256) {
        int c = i % NC, w = i / NC;
        sx[c * 26 + w] = xsrc[i] - smean[c];
    }
    for (int i = tid; i < (NCP - NC) * NW; i += 256) {
        int c = NC + i / NW, w = i % NW;
        sx[c * 26 + w] = 0.f;
    }
    __syncthreads();

    int wv = tid >> 5, lane = tid & 31, half = lane >> 4, ln = lane & 15;

    // B fragments (shared across all row tiles): B[w][k] = M[k][w]
    v2f bf[6][2];
    for (int kk = 0; kk < 6; ++kk) {
        int w0 = kk * 4 + 2 * half;
        for (int nt = 0; nt < 2; ++nt) {
            int n = nt * 16 + ln;
            v2f t; t.x = 0.f; t.y = 0.f;
            if (n < NW) { t.x = sM[n * NW + w0]; t.y = sM[n * NW + w0 + 1]; }
            bf[kk][nt] = t;
        }
    }

    for (int tile = wv; tile < 21; tile += 8) {      // 21*16 == 336 padded channels
        int c0 = tile * 16;
        v8f acc0 = {}; v8f acc1 = {};
        for (int kk = 0; kk < 6; ++kk) {             // K = 24 in 6 steps of 4
            int w0 = kk * 4 + 2 * half;
            v2f a = lds_v2(&sx[(c0 + ln) * 26 + w0]);   // unconditional, aligned b64
            acc0 = wmma4(a, bf[kk][0], acc0);
            acc1 = wmma4(a, bf[kk][1], acc1);
        }
        if (c0 + 15 < NC) {
            // fast path: all 16 channel rows valid (20 of 21 tiles)
#pragma unroll
            for (int r = 0; r < 8; ++r) {
                int c = c0 + r + 8 * half;
                size_t base = (((size_t)b * NC + c) * NS + s) * NW;
                xd[base + ln] = acc0[r];
                int k1 = 16 + ln;
                if (k1 < NW) xd[base + k1] = acc1[r];
            }
        } else {
            for (int r = 0; r < 8; ++r) {
                int c = c0 + r + 8 * half;
                if (c >= NC) continue;
                size_t base = (((size_t)b * NC + c) * NS + s) * NW;
                xd[base + ln] = acc0[r];
                int k1 = 16 + ln;
                if (k1 < NW) xd[base + k1] = acc1[r];
            }
        }
    }
}

// ---------------- K3/K4: depthwise 24x24 conv (edge pad) + per-ch scale/bias (+opt residual) ----
// block = 128 threads; 120 active compute threads: s = t/4, w0 = (t%4)*6 (6 outputs each)
__global__ void k_conv(const float* __restrict__ in, const float* __restrict__ dww,
                       const float* __restrict__ dwb, const float* __restrict__ pww,
                       const float* __restrict__ pwb, const float* __restrict__ res,
                       float* __restrict__ out) {
    __shared__ __align__(16) float imgraw[NS * NW];     // 720 floats = 180 x b128
    __shared__ __align__(16) float imgp[53 * 48];       // edge-padded (53 x 47, stride 48)
    __shared__ __align__(16) float ker[NW * NW];        // 576 floats = 144 x b128
    int bc = blockIdx.x;            // b*321 + c
    int c  = bc % NC;
    int tid = threadIdx.x;

    // async direct-to-LDS staging (no VGPR round trip)
    const float* src = in + (size_t)bc * (NS * NW);
    for (int i = tid; i < 180; i += 128) async_ld16(&imgraw[i * 4], src + i * 4);
    const float* kw = dww + (size_t)c * (NW * NW);
    for (int i = tid; i < 144; i += 128) async_ld16(&ker[i * 4], kw + i * 4);
    wait_async0();
    __syncthreads();

    // build padded image: imgp[sp][wp] = imgraw[clamp(sp-11)][clamp(wp-11)]
    for (int i = tid; i < 53 * 47; i += 128) {
        int sp = i / 47, wp = i % 47;
        int si = sp - 11; si = si < 0 ? 0 : (si > NS - 1 ? NS - 1 : si);
        int wj = wp - 11; wj = wj < 0 ? 0 : (wj > NW - 1 ? NW - 1 : wj);
        imgp[sp * 48 + wp] = imgraw[si * NW + wj];
    }
    __syncthreads();

    if (tid < 120) {
        int s  = tid >> 2;
        int w0 = (tid & 3) * 6;
        float db = dwb[c], pw = pww[c], pb = pwb[c];
        float acc[6];
#pragma unroll
        for (int d = 0; d < 6; ++d) acc[d] = db;
        for (int i = 0; i < NW; ++i) {
            const float* rp = &imgp[(s + i) * 48 + w0];
            const float* kr = &ker[i * NW];
#pragma unroll
            for (int j = 0; j < NW; ++j) {
                float kv = kr[j];                 // wave-uniform -> LDS broadcast
#pragma unroll
                for (int d = 0; d < 6; ++d)
                    acc[d] = fmaf(rp[j + d], kv, acc[d]);   // sliding window, CSE'd
            }
        }
        size_t base = (size_t)bc * (NS * NW) + s * NW + w0;
#pragma unroll
        for (int d = 0; d < 6; ++d) {
            float r = acc[d] * pw + pb;
            if (res) r += res[base + d];
            out[base + d] = r;
        }
    }
}

// ---------------- K5: per-channel FFN via WMMA: [384x30]@W1[30x16] -> +b1 -> @W2[16x30] -> +b2 ----
__global__ void k_ffn(const float* __restrict__ xd, const float* __restrict__ W1,
                      const float* __restrict__ b1, const float* __restrict__ W2,
                      const float* __restrict__ b2, float* __restrict__ out0) {
    __shared__ __align__(16) float sA[384 * 34];    // rows=(b_local*24+w), cols=s (30->32 pad, stride 34)
    __shared__ __align__(16) float sW1[32 * 16];    // [s][h], s padded to 32
    __shared__ __align__(16) float sW2[16 * 32];    // [h][p], p padded to 32
    __shared__ __align__(16) float sh[8 * 16 * 17]; // per-wave 16x16 D->A transpose
    __shared__ float sb1[16];
    __shared__ float sb2[32];

    int bid = blockIdx.x;
    int c = bid % NC, bg = bid / NC;     // 4 groups of 16 batches
    int tid = threadIdx.x;

    for (int i = tid; i < 384 * 32; i += 256) {
        int row = i >> 5, scol = i & 31;
        int bl = row / NW, w = row % NW;
        float v = 0.f;
        if (scol < NS)
            v = xd[(((size_t)(bg * 16 + bl) * NC + c) * NS + scol) * NW + w];
        sA[row * 34 + scol] = v;
    }
    for (int i = tid; i < 512; i += 256) {
        int s = i >> 4, h = i & 15;
        sW1[i] = (s < NS) ? W1[((size_t)c * NS + s) * NHID + h] : 0.f;
    }
    for (int i = tid; i < 512; i += 256) {
        int h = i >> 5, p = i & 31;
        sW2[i] = (p < NS) ? W2[((size_t)c * NHID + h) * NS + p] : 0.f;
    }
    if (tid < 16) sb1[tid] = b1[c * NHID + tid];
    if (tid < 32) sb2[tid] = (tid < NS) ? b2[c * NS + tid] : 0.f;
    __syncthreads();

    int wv = tid >> 5, lane = tid & 31, half = lane >> 4, ln = lane & 15;

    v2f bf1[8];                           // GEMM1 B frags: K=32 (padded)
    for (int kk = 0; kk < 8; ++kk) {
        int s0 = kk * 4 + 2 * half;
        v2f t; t.x = sW1[s0 * 16 + ln]; t.y = sW1[(s0 + 1) * 16 + ln];
        bf1[kk] = t;
    }
    v2f bf2[4][2];                        // GEMM2 B frags: K=16, N=30 (2 tiles)
    for (int kk = 0; kk < 4; ++kk) {
        int h0 = kk * 4 + 2 * half;
        for (int nt = 0; nt < 2; ++nt) {
            int p = nt * 16 + ln;
            v2f t; t.x = sW2[h0 * 32 + p]; t.y = sW2[(h0 + 1) * 32 + p];
            bf2[kk][nt] = t;
        }
    }

    float* hme = &sh[wv * 16 * 17];
    for (int tile = wv; tile < 24; tile += 8) {   // 384 rows / 16
        int r0 = tile * 16;
        v8f acc = {};
        for (int kk = 0; kk < 8; ++kk) {
            int s0 = kk * 4 + 2 * half;
            v2f a = lds_v2(&sA[(r0 + ln) * 34 + s0]);   // aligned b64
            acc = wmma4(a, bf1[kk], acc);
        }
        // D-layout -> A-layout transpose through LDS, fused +b1
        for (int r = 0; r < 8; ++r) {
            int m = r + 8 * half;
            hme[m * 17 + ln] = acc[r] + sb1[ln];
        }
        v8f o0 = {}, o1 = {};
        for (int kk = 0; kk < 4; ++kk) {
            int h0 = kk * 4 + 2 * half;
            v2f a; a.x = hme[ln * 17 + h0]; a.y = hme[ln * 17 + h0 + 1];
            o0 = wmma4(a, bf2[kk][0], o0);
            o1 = wmma4(a, bf2[kk][1], o1);
        }
        for (int r = 0; r < 8; ++r) {
            int row = r0 + r + 8 * half;
            int bl = row / NW, w = row % NW;
            size_t base = ((size_t)(bg * 16 + bl) * NC + c) * NS;
            out0[(base + ln) * NW + w] = o0[r] + sb2[ln];
            int p1 = 16 + ln;
            if (p1 < NS) out0[(base + p1) * NW + w] = o1[r] + sb2[p1];
        }
    }
}

// ---------------- K6: IDCT via WMMA + transpose to [B,720,C] + mean add ----------------
__global__ void k_idct(const float* __restrict__ out0, const float* __restrict__ Mp,
                       const float* __restrict__ meanp, float* __restrict__ dout) {
    __shared__ __align__(16) float sA[128 * 28];   // 128 rows x K=24, stride 28 (16B-aligned rows)
    __shared__ __align__(16) float sM[NW * NW];
    int bid = blockIdx.x;
    size_t row0 = (size_t)bid * 128;
    int tid = threadIdx.x;

    // async stage 128x24 A tile: 6 aligned b128 per row (24 % 4 == 0, stride 28 -> 16B aligned)
    const float* src = out0 + row0 * NW;
    for (int i = tid; i < 128 * 6; i += 256) {
        int row = i / 6, q = i % 6;
        async_ld16(&sA[row * 28 + q * 4], src + i * 4);
    }
    for (int i = tid; i < NW * NW; i += 256) sM[i] = Mp[i];
    wait_async0();
    __syncthreads();

    int wv = tid >> 5, lane = tid & 31, half = lane >> 4, ln = lane & 15;

    v2f bf[6][2];                    // B[k][w] = M[k][w]
    for (int kk = 0; kk < 6; ++kk) {
        int k0 = kk * 4 + 2 * half;
        for (int nt = 0; nt < 2; ++nt) {
            int n = nt * 16 + ln;
            v2f t; t.x = 0.f; t.y = 0.f;
            if (n < NW) { t.x = sM[k0 * NW + n]; t.y = sM[(k0 + 1) * NW + n]; }
            bf[kk][nt] = t;
        }
    }

    int r0 = wv * 16;                // one 16-row tile per wave
    v8f a0 = {}, a1 = {};
    for (int kk = 0; kk < 6; ++kk) {
        int k0 = kk * 4 + 2 * half;
        v2f a = lds_v2(&sA[(r0 + ln) * 28 + k0]);   // aligned b64
        a0 = wmma4(a, bf[kk][0], a0);
        a1 = wmma4(a, bf[kk][1], a1);
    }
    for (int r = 0; r < 8; ++r) {
        size_t row = row0 + (size_t)(r0 + r + 8 * half);   // (b*321+c)*30 + p
        size_t bc = row / NS; int p = (int)(row % NS);
        int b = (int)(bc / NC), c = (int)(bc % NC);
        float mv = meanp[bc];
        dout[((size_t)b * SEQ + p * NW + ln) * NC + c] = a0[r] + mv;
        int w1 = 16 + ln;
        if (w1 < NW) dout[((size_t)b * SEQ + p * NW + w1) * NC + c] = a1[r] + mv;
    }
}

// ---------------- launch ----------------
extern "C" void kernel_launch(void* const* d_in, const int* in_sizes, int n_in,
                              void* d_out, int out_size, void* d_ws, size_t ws_size,
                              hipStream_t stream) {
    const float* x    = (const float*)d_in[0];
    const float* dw1w = (const float*)d_in[1];
    const float* dw1b = (const float*)d_in[2];
    const float* pw1w = (const float*)d_in[3];
    const float* pw1b = (const float*)d_in[4];
    const float* dw2w = (const float*)d_in[5];
    const float* dw2b = (const float*)d_in[6];
    const float* pw2w = (const float*)d_in[7];
    const float* pw2b = (const float*)d_in[8];
    const float* W1   = (const float*)d_in[9];
    const float* b1   = (const float*)d_in[10];
    const float* W2   = (const float*)d_in[11];
    const float* b2   = (const float*)d_in[12];
    float* out = (float*)d_out;

    float* ws    = (float*)d_ws;
    float* Mp    = ws;                                  // 576 floats
    float* meanp = ws + 1024;                           // 20544 floats
    float* xd    = ws + 32768;                          // 14,791,680 floats
    float* tmp   = xd + (size_t)NB * NC * NS * NW;      // 14,791,680 floats

    k_initM<<<1, 256, 0, stream>>>(Mp);
    k_mean <<<NB, 352, 0, stream>>>(x, meanp);
    k_dct  <<<NB * NS, 256, 0, stream>>>(x, Mp, meanp, xd);
    k_conv <<<NB * NC, 128, 0, stream>>>(xd, dw1w, dw1b, pw1w, pw1b, nullptr, tmp);
    k_conv <<<NB * NC, 128, 0, stream>>>(tmp, dw2w, dw2b, pw2w, pw2b, xd, xd);
    k_ffn  <<<NC * 4, 256, 0, stream>>>(xd, W1, b1, W2, b2, tmp);
    k_idct <<<(NB * NC * NS) / 128, 256, 0, stream>>>(tmp, Mp, meanp, out);
}